// MPLayer_1889785610809
// MI455X (gfx1250) — compile-verified
//
#include <hip/hip_runtime.h>
#include <math.h>

typedef __attribute__((ext_vector_type(2))) float v2f;
typedef __attribute__((ext_vector_type(8))) float v8f;

// ---------------------------------------------------------------------------
// Phase 1: h = semantics[:, 0] @ W   (n x 128) = (n x 128) * (128 x 128), f32
// One wave computes one 16x16 output tile via V_WMMA_F32_16X16X4_F32.
// Block = 256 threads = 8 waves -> one 16-row M block x full N=128.
// semantics has shape (n, 4, 128); row stride of the [:,0] slice is 512 floats.
// ---------------------------------------------------------------------------
__global__ __launch_bounds__(256) void mp_gemm_wmma(
    const float* __restrict__ sem,   // (n, 4, 128)
    const float* __restrict__ W,     // (128, 128) row-major
    float* __restrict__ h,           // (n, 128) output (staged in d_out)
    int n)
{
    const int lane  = threadIdx.x & 31;
    const int wave  = threadIdx.x >> 5;
    const int m0    = blockIdx.x * 16;
    const int n0    = wave * 16;
    const int col   = lane & 15;          // M-row (for A) / N-col (for B, C/D)
    const int khalf = (lane >> 4) * 2;    // lanes 0-15 -> K {k,k+1}; 16-31 -> {k+2,k+3}

    // Clamp A-row reads for the M tail; WMMA requires EXEC all-ones, so we
    // never predicate around the WMMA itself.
    int arow_idx = m0 + col;
    if (arow_idx > n - 1) arow_idx = n - 1;
    const float* __restrict__ arow = sem + (size_t)arow_idx * 512; // [:,0] slice

    v8f acc = {};
    #pragma unroll 4
    for (int k = 0; k < 128; k += 4) {
        v2f a, b;
        a = *(const v2f*)(arow + k + khalf);            // one b64 load
        // B fragment: row K striped across lanes within a VGPR.
        b.x = W[(size_t)(k + khalf)     * 128 + n0 + col];
        b.y = W[(size_t)(k + khalf + 1) * 128 + n0 + col];
        acc = __builtin_amdgcn_wmma_f32_16x16x4_f32(
            /*neg_a=*/false, a, /*neg_b=*/false, b,
            /*c_mod=*/(short)0, acc, /*reuse_a=*/false, /*reuse_b=*/false);
    }

    // C/D layout: VGPR r -> (M = r, N = lane) for lanes 0-15,
    //                        (M = r+8, N = lane-16) for lanes 16-31.
    const int mbase = m0 + (lane >> 4) * 8;
    float* __restrict__ od = h + (size_t)mbase * 128 + n0 + col;
    if (m0 + 16 <= n) {
        // Fast path (all blocks when n % 16 == 0): no per-row guards, no
        // exec-mask cascade in the epilogue.
        #pragma unroll
        for (int r = 0; r < 8; ++r)
            od[(size_t)r * 128] = acc[r];
    } else {
        #pragma unroll
        for (int r = 0; r < 8; ++r)
            if (mbase + r < n)
                od[(size_t)r * 128] = acc[r];
    }
}

// ---------------------------------------------------------------------------
// Phase 2: segment-sum scatter. One wave per edge; each lane moves 4 floats
// (b128 gather out of L2-resident h, 4x global_atomic_add_f32 into sums).
// ---------------------------------------------------------------------------
__global__ __launch_bounds__(256) void mp_edge_scatter(
    const float* __restrict__ h,         // (n, 128)
    const int* __restrict__ edge_index,  // (n_edges, 2) interleaved [src, tgt]
    float* __restrict__ sums,            // (n, 128)
    float* __restrict__ cnt,             // (n,)
    int n_edges)
{
    const int edge = blockIdx.x * 8 + (threadIdx.x >> 5);
    if (edge >= n_edges) return;
    const int lane = threadIdx.x & 31;

    const int src = edge_index[(size_t)edge * 2 + 0];
    const int tgt = edge_index[(size_t)edge * 2 + 1];

    const float4 v = ((const float4*)(h + (size_t)src * 128))[lane];
    float* sb = sums + (size_t)tgt * 128 + lane * 4;
    atomicAdd(sb + 0, v.x);
    atomicAdd(sb + 1, v.y);
    atomicAdd(sb + 2, v.z);
    atomicAdd(sb + 3, v.w);
    if (lane == 0)
        atomicAdd(cnt + tgt, 1.0f);
}

// ---------------------------------------------------------------------------
// Phase 3: out = gelu_exact(sums / max(cnt, 1)), vectorized float4.
// Each thread handles 4 consecutive columns of one row (hidden=128 -> 32
// threads per row), so b128 load/store and a single cnt load per thread.
// ---------------------------------------------------------------------------
__global__ __launch_bounds__(256) void mp_finalize(
    const float* __restrict__ sums,
    const float* __restrict__ cnt,
    float* __restrict__ out,
    int n)
{
    const size_t q = (size_t)blockIdx.x * blockDim.x + threadIdx.x; // float4 idx
    const size_t total4 = (size_t)n * 32;                           // 128/4 per row
    if (q >= total4) return;
    const int row = (int)(q >> 5);
    const float inv = 1.0f / fmaxf(cnt[row], 1.0f);

    float4 s = ((const float4*)sums)[q];
    float4 r;
    const float k = 0.70710678118654752f;
    r.x = 0.5f * (s.x * inv) * (1.0f + erff(s.x * inv * k));
    r.y = 0.5f * (s.y * inv) * (1.0f + erff(s.y * inv * k));
    r.z = 0.5f * (s.z * inv) * (1.0f + erff(s.z * inv * k));
    r.w = 0.5f * (s.w * inv) * (1.0f + erff(s.w * inv * k));
    ((float4*)out)[q] = r;
}

extern "C" void kernel_launch(void* const* d_in, const int* in_sizes, int n_in,
                              void* d_out, int out_size, void* d_ws, size_t ws_size,
                              hipStream_t stream) {
    const float* sem        = (const float*)d_in[0];  // (n, 4, 128) f32
    // d_in[1] = attention_masks, unused by the reference computation
    const float* W          = (const float*)d_in[2];  // (128, 128) f32
    const int*   edge_index = (const int*)d_in[3];    // (n_edges, 2) int

    const int n       = in_sizes[0] / (4 * 128);
    const int n_edges = in_sizes[3] / 2;

    float* out  = (float*)d_out;            // stages h, then holds final gelu
    float* sums = (float*)d_ws;             // (n, 128)
    float* cnt  = sums + (size_t)n * 128;   // (n,)

    // Zero the accumulators (graph-capture-safe stream op).
    hipMemsetAsync(d_ws, 0, ((size_t)n * 128 + (size_t)n) * sizeof(float), stream);

    // Phase 1: GEMM via f32 WMMA, h staged into d_out.
    {
        dim3 grid((n + 15) / 16);
        mp_gemm_wmma<<<grid, 256, 0, stream>>>(sem, W, out, n);
    }

    // Phase 2: edge scatter (8 edges per 256-thread block).
    {
        dim3 grid((n_edges + 7) / 8);
        mp_edge_scatter<<<grid, 256, 0, stream>>>(out, edge_index, sums, cnt, n_edges);
    }

    // Phase 3: mean + exact GELU, overwrite d_out with final result.
    {
        size_t total4 = (size_t)n * 32;
        dim3 grid((unsigned)((total4 + 255) / 256));
        mp_finalize<<<grid, 256, 0, stream>>>(sums, cnt, out, n);
    }
}